// SteerableGeometricProductLayer_28046136443384
// MI455X (gfx1250) — compile-verified
//
#include <hip/hip_runtime.h>
#include <hip/hip_bf16.h>
#include <math.h>

typedef float v2f __attribute__((ext_vector_type(2)));
typedef float v8f __attribute__((ext_vector_type(8)));

namespace {
constexpr int kB  = 8192;
constexpr int kF  = 512;
constexpr int kD  = 8;
constexpr int kNG = 4;
constexpr int BM  = 32;
constexpr int BN  = 32;

__device__ __forceinline__ v2f mk2(float a, float b) { v2f r; r.x = a; r.y = b; return r; }

// Sign of blade product a*b in Cl(3,0): (-1)^(sum_t popc((a>>t)&b)), metric all +1.
__device__ __forceinline__ float gp_sign(int a, int b) {
  const int c = __popc((a >> 1) & b) + __popc((a >> 2) & b);
  return (c & 1) ? -1.0f : 1.0f;
}
} // namespace

__global__ __launch_bounds__(256)
void geoprod_fused(const float* __restrict__ x,
                   const float* __restrict__ w_gp,
                   const float* __restrict__ w_right,
                   const float* __restrict__ a_norm,
                   const float* __restrict__ w_lin,
                   const float* __restrict__ b_lin,
                   float* __restrict__ out)
{
  __shared__ float lds[2][kD][BM][BN];   // 64 KB: [set][blade][m][n]

  constexpr int GRADE[kD] = {0,1,1,1,2,2,2,3};
  constexpr int MASKP[kD] = {0,1,2,4,3,5,6,7};   // pos<->mask (involution)
  // pid[gi][gj][gk], row-major numbering of valid grade paths (P=20)
  constexpr int PID[4][4][4] = {
    { {0,0,0,0},  {0,1,0,0},  {0,0,2,0},  {0,0,0,3}  },
    { {0,4,0,0},  {5,0,6,0},  {0,7,0,8},  {0,0,9,0}  },
    { {0,0,10,0}, {0,11,0,12},{13,0,14,0},{0,15,0,0} },
    { {0,0,0,16}, {0,0,17,0}, {0,18,0,0}, {19,0,0,0} }
  };

  const int nn0  = blockIdx.x * BN;
  const int bm0  = blockIdx.y * BM;
  const int tid  = threadIdx.x;
  const int lane = tid & 31;
  const int wv   = tid >> 5;       // 0..7
  const int set  = wv & 1;         // 0: w_right, 1: w_lin
  const int bp   = wv >> 1;        // blade pair 0..3 -> blades 2bp, 2bp+1
  const int i0   = bp << 1;
  const int g0   = GRADE[i0];
  const int g1   = GRADE[i0 + 1];

  const float* __restrict__ W = set ? w_lin : w_right;

  const int lrow  = lane & 15;
  const int khalf = (lane >> 4) << 1;   // K offset 0 (lanes 0-15) or 2 (lanes 16-31)

  v8f acc[2][2][2];                      // [blade][mi][ni]
  #pragma unroll
  for (int ib = 0; ib < 2; ++ib)
    #pragma unroll
    for (int mi = 0; mi < 2; ++mi)
      #pragma unroll
      for (int ni = 0; ni < 2; ++ni)
        #pragma unroll
        for (int r = 0; r < 8; ++r) acc[ib][mi][ni][r] = 0.0f;

  const float* ax0 = x + ((size_t)(bm0 + lrow)      * kF) * kD + i0;
  const float* ax1 = x + ((size_t)(bm0 + 16 + lrow) * kF) * kD + i0;
  const float* bw0 = W + ((size_t)(nn0 + lrow)      * kF) * kNG;
  const float* bw1 = W + ((size_t)(nn0 + 16 + lrow) * kF) * kNG;

  #pragma unroll 2
  for (int k0 = 0; k0 < kF; k0 += 4) {
    const int k = k0 + khalf;
    // A: one float2 covers both blades of the pair (adjacent, 8B aligned)
    const v2f p00 = *(const v2f*)(ax0 + (size_t)k * kD);
    const v2f p01 = *(const v2f*)(ax0 + (size_t)(k + 1) * kD);
    const v2f p10 = *(const v2f*)(ax1 + (size_t)k * kD);
    const v2f p11 = *(const v2f*)(ax1 + (size_t)(k + 1) * kD);

    v2f aF[2][2];                        // [blade][mi]; comp0=K(khalf), comp1=K+1
    aF[0][0] = mk2(p00.x, p01.x);
    aF[1][0] = mk2(p00.y, p01.y);
    aF[0][1] = mk2(p10.x, p11.x);
    aF[1][1] = mk2(p10.y, p11.y);

    v2f bF[2][2];                        // [blade(grade)][ni]; B[k][n] = W[n,k,g]
    bF[0][0] = mk2(bw0[(size_t)k * kNG + g0], bw0[(size_t)(k + 1) * kNG + g0]);
    bF[1][0] = mk2(bw0[(size_t)k * kNG + g1], bw0[(size_t)(k + 1) * kNG + g1]);
    bF[0][1] = mk2(bw1[(size_t)k * kNG + g0], bw1[(size_t)(k + 1) * kNG + g0]);
    bF[1][1] = mk2(bw1[(size_t)k * kNG + g1], bw1[(size_t)(k + 1) * kNG + g1]);

    #pragma unroll
    for (int ib = 0; ib < 2; ++ib)
      #pragma unroll
      for (int mi = 0; mi < 2; ++mi)
        #pragma unroll
        for (int ni = 0; ni < 2; ++ni)
          acc[ib][mi][ni] = __builtin_amdgcn_wmma_f32_16x16x4_f32(
              false, aF[ib][mi], false, bF[ib][ni], (short)0,
              acc[ib][mi][ni], false, false);
  }

  // Spill accumulators to LDS: C layout = lane holds N=lane&15, rows r (+8 for hi half)
  #pragma unroll
  for (int ib = 0; ib < 2; ++ib)
    #pragma unroll
    for (int mi = 0; mi < 2; ++mi)
      #pragma unroll
      for (int ni = 0; ni < 2; ++ni)
        #pragma unroll
        for (int r = 0; r < 8; ++r) {
          const int M = mi * 16 + r + ((lane >> 4) << 3);
          const int N = ni * 16 + (lane & 15);
          lds[set][i0 + ib][M][N] = acc[ib][mi][ni][r];
        }
  __syncthreads();

  // -------- elementwise epilogue: norms, gate, sparse geometric product --------
  const int n_local = tid & (BN - 1);
  const int bgrp    = tid >> 5;
  const int n       = nn0 + n_local;

  float wg[20];
  #pragma unroll
  for (int p = 0; p < 20; ++p) wg[p] = w_gp[(size_t)n * 20 + p];
  float sg[4];
  #pragma unroll
  for (int g = 0; g < 4; ++g) {
    const float a = a_norm[(size_t)n * 4 + g];
    sg[g] = 1.0f / (1.0f + expf(-a));
  }
  const float bl = b_lin[n];
  constexpr float RS2 = 0.70710678118654752f;

  #pragma unroll
  for (int eb = 0; eb < 4; ++eb) {
    const int b_local = bgrp * 4 + eb;
    const int b = bm0 + b_local;

    float xr[8], li[8], xv[8];
    #pragma unroll
    for (int i = 0; i < 8; ++i) {
      xr[i] = lds[0][i][b_local][n_local];
      li[i] = lds[1][i][b_local][n_local];
    }
    const float4 q0 = *(const float4*)(x + ((size_t)b * kF + n) * kD);
    const float4 q1 = *(const float4*)(x + ((size_t)b * kF + n) * kD + 4);
    xv[0]=q0.x; xv[1]=q0.y; xv[2]=q0.z; xv[3]=q0.w;
    xv[4]=q1.x; xv[5]=q1.y; xv[6]=q1.z; xv[7]=q1.w;

    float nrm[4];
    nrm[0] = sqrtf(xr[0]*xr[0]);
    nrm[1] = sqrtf(xr[1]*xr[1] + xr[2]*xr[2] + xr[3]*xr[3]);
    nrm[2] = sqrtf(xr[4]*xr[4] + xr[5]*xr[5] + xr[6]*xr[6]);
    nrm[3] = sqrtf(xr[7]*xr[7]);

    float den[4];
    #pragma unroll
    for (int g = 0; g < 4; ++g)
      den[g] = sg[g] * (nrm[g] - 1.0f) + 1.0f + 1e-6f;

    float xrn[8];
    #pragma unroll
    for (int i = 0; i < 8; ++i) xrn[i] = xr[i] / den[GRADE[i]];

    float res[8];
    #pragma unroll
    for (int j = 0; j < 8; ++j) res[j] = li[j];
    res[0] += bl;   // bias on scalar blade only

    // out_j = sum_{i,k} sign(i,k) * w_gp[n, pid(gi,gj,gk)] * x_i * xrn_k, j = pos(mask_i^mask_k)
    #pragma unroll
    for (int i = 0; i < 8; ++i) {
      #pragma unroll
      for (int kk = 0; kk < 8; ++kk) {
        const int ma = MASKP[i];
        const int mb = MASKP[kk];
        const int j  = MASKP[ma ^ mb];
        const float s = gp_sign(ma, mb);
        const int  p = PID[GRADE[i]][GRADE[j]][GRADE[kk]];
        res[j] += s * wg[p] * xv[i] * xrn[kk];
      }
    }

    float4 o0, o1;
    o0.x = res[0]*RS2; o0.y = res[1]*RS2; o0.z = res[2]*RS2; o0.w = res[3]*RS2;
    o1.x = res[4]*RS2; o1.y = res[5]*RS2; o1.z = res[6]*RS2; o1.w = res[7]*RS2;
    float* op = out + ((size_t)b * kF + n) * kD;
    *(float4*)op       = o0;
    *(float4*)(op + 4) = o1;
  }
}

extern "C" void kernel_launch(void* const* d_in, const int* in_sizes, int n_in,
                              void* d_out, int out_size, void* d_ws, size_t ws_size,
                              hipStream_t stream) {
  (void)in_sizes; (void)n_in; (void)out_size; (void)d_ws; (void)ws_size;
  const float* x       = (const float*)d_in[0];
  const float* w_gp    = (const float*)d_in[1];
  const float* w_right = (const float*)d_in[2];
  const float* a_norm  = (const float*)d_in[3];
  const float* w_lin   = (const float*)d_in[4];
  const float* b_lin   = (const float*)d_in[5];

  dim3 grid(kF / BN, kB / BM);   // 16 x 256 blocks
  geoprod_fused<<<grid, 256, 0, stream>>>(x, w_gp, w_right, a_norm, w_lin, b_lin,
                                          (float*)d_out);
}